// DGLGATNE_6923487281349
// MI455X (gfx1250) — compile-verified
//
#include <hip/hip_runtime.h>
#include <math.h>

// Problem constants (match reference)
#define NUM_NODES 500000
#define DD        128
#define UU        32
#define TT        4
#define AA        32
#define N_SRC     65536
#define BB        8192
#define EE        262144
#define EPSF      1e-12f

typedef __attribute__((ext_vector_type(2))) float v2f;
typedef __attribute__((ext_vector_type(8))) float v8f;

// ---------------------------------------------------------------------------
// Kernel 1: edge scatter-add.  One wave32 per edge, lane = u (0..31).
//   agg[t][dst][u] += node_type_embeddings[input_nodes[edge_src[t,e]]][t][u]
// Gather is a coalesced 128B row per wave; atomics land in the 4MB
// L2-resident aggregate.
// ---------------------------------------------------------------------------
__global__ void dglgatne_scatter(const int* __restrict__ edge_src,
                                 const int* __restrict__ edge_dst,
                                 const int* __restrict__ input_nodes,
                                 const float* __restrict__ nte,   // (NUM_NODES,T,U)
                                 float* __restrict__ agg)          // (T,B,U)
{
    const int lane = threadIdx.x & 31;
    const int wave = threadIdx.x >> 5;
    const int t    = blockIdx.y;
    const int e    = blockIdx.x * 8 + wave;   // 8 waves / block

    const int src  = edge_src[t * EE + e];
    const int dst  = edge_dst[t * EE + e];
    const int node = input_nodes[src];

    const float v = nte[((size_t)node * TT + t) * UU + lane];
    atomicAdd(&agg[((size_t)t * BB + dst) * UU + lane], v);
}

// ---------------------------------------------------------------------------
// Kernel 2: per-row attention.  One wave per b-row (8 waves / block).
//   h[t,a]   = tanh(sum_u nte[b,t,u] * w_s1[t,u,a])      (lane = a)
//   score[t] = sum_a h[t,a]*w_s2[t,a]                    (cross-lane reduce)
//   att      = softmax_t(score);  combined[b,u] = sum_t att[t]*nte[b,t,u]
// ---------------------------------------------------------------------------
__global__ void dglgatne_attention(const float* __restrict__ agg,   // (T,B,U)
                                   const float* __restrict__ w_s1,  // (T,U,A)
                                   const float* __restrict__ w_s2,  // (T,A)
                                   float* __restrict__ combined)    // (B,U)
{
    __shared__ float lds_w1[TT * UU * AA];   // 16 KB
    __shared__ float lds_w2[TT * AA];        // 512 B

    const int tid  = threadIdx.x;
    const int lane = tid & 31;
    const int wave = tid >> 5;
    const int b    = blockIdx.x * 8 + wave;

    for (int i = tid; i < TT * UU * AA; i += 256) lds_w1[i] = w_s1[i];
    for (int i = tid; i < TT * AA;      i += 256) lds_w2[i] = w_s2[i];
    __syncthreads();

    float n[TT];
#pragma unroll
    for (int t = 0; t < TT; ++t)
        n[t] = agg[((size_t)t * BB + b) * UU + lane];

    float score[TT];
#pragma unroll
    for (int t = 0; t < TT; ++t) {
        float h = 0.0f;
#pragma unroll
        for (int u = 0; u < UU; ++u) {
            const float nv = __shfl(n[t], u, 32);       // broadcast nte[b,t,u]
            h = fmaf(nv, lds_w1[(t * UU + u) * AA + lane], h);
        }
        h = tanhf(h);
        float p = h * lds_w2[t * AA + lane];
        // full-wave sum (wave32)
        p += __shfl_xor(p, 16, 32);
        p += __shfl_xor(p, 8, 32);
        p += __shfl_xor(p, 4, 32);
        p += __shfl_xor(p, 2, 32);
        p += __shfl_xor(p, 1, 32);
        score[t] = p;
    }

    // softmax over T=4 (identical on every lane)
    float m = score[0];
#pragma unroll
    for (int t = 1; t < TT; ++t) m = fmaxf(m, score[t]);
    float ex[TT], s = 0.0f;
#pragma unroll
    for (int t = 0; t < TT; ++t) { ex[t] = expf(score[t] - m); s += ex[t]; }
    const float inv = 1.0f / s;

    float c = 0.0f;
#pragma unroll
    for (int t = 0; t < TT; ++t) c = fmaf(ex[t] * inv, n[t], c);

    combined[(size_t)b * UU + lane] = c;
}

// ---------------------------------------------------------------------------
// Kernel 3: final GEMM (WMMA f32 16x16x4) + embedding add + L2 normalize.
// Block = 8 waves = one 16-row b-tile for one type t; wave w owns d-tile
// d0 = 16*w.  K = U = 32 -> 8 WMMA steps.
//
// V_WMMA_F32_16X16X4_F32 lane layouts (ISA 7.12.2):
//   A (16x4): M = lane&15; VGPR0 holds K = 2*(lane>>4), VGPR1 holds K+1
//   B (4x16): N = lane&15; VGPR0 holds K = 2*(lane>>4), VGPR1 holds K+1
//   C/D     : N = lane&15; VGPR r holds M = r + 8*(lane>>4)
// ---------------------------------------------------------------------------
__global__ void dglgatne_gemm_norm(const float* __restrict__ combined, // (B,U)
                                   const float* __restrict__ trans_w,  // (T,U,D)
                                   const float* __restrict__ node_emb, // (NUM_NODES,D)
                                   const int* __restrict__ out_nodes,  // (B)
                                   float* __restrict__ out)            // (B,T,D)
{
    __shared__ float lds_emb[16 * DD];   // 8 KB: gathered embedding tile
    __shared__ int   lds_on[16];
    __shared__ float lds_ss[16];         // per-row sum of squares

    const int tid  = threadIdx.x;
    const int lane = tid & 31;
    const int wave = tid >> 5;
    const int col  = lane & 15;          // N index within tile
    const int half = lane >> 4;

    const int b0 = blockIdx.x * 16;
    const int t  = blockIdx.y;
    const int d0 = wave * 16;

    if (tid < 16) {
        lds_on[tid] = out_nodes[b0 + tid];
        lds_ss[tid] = 0.0f;
    }
    __syncthreads();

    // coalesced stage of the 16 gathered embedding rows
    for (int i = tid; i < 16 * DD; i += 256) {
        const int r = i >> 7, c = i & (DD - 1);
        lds_emb[i] = node_emb[(size_t)lds_on[r] * DD + c];
    }

    // ---- WMMA accumulation over K = 32 ----
    v8f acc = {0.f, 0.f, 0.f, 0.f, 0.f, 0.f, 0.f, 0.f};
    const float* arow = combined + (size_t)(b0 + col) * UU;   // A: M = col(lane&15)
    const float* bmat = trans_w + (size_t)t * UU * DD;        // B[k][n] = trans_w[t][k][d0+n]
#pragma unroll
    for (int kk = 0; kk < 8; ++kk) {
        const int k = kk * 4 + half * 2;
        v2f a, b;
        a.x = arow[k];
        a.y = arow[k + 1];
        b.x = bmat[(size_t)k * DD + d0 + col];
        b.y = bmat[(size_t)(k + 1) * DD + d0 + col];
        acc = __builtin_amdgcn_wmma_f32_16x16x4_f32(
                /*neg_a=*/false, a, /*neg_b=*/false, b,
                /*c_mod=*/(short)0, acc, /*reuse_a=*/false, /*reuse_b=*/false);
    }
    __syncthreads();   // lds_emb ready

    // ---- add gathered embedding, build per-row sum of squares ----
    float val[8];
#pragma unroll
    for (int r = 0; r < 8; ++r) {
        const int mrow = r + 8 * half;
        val[r] = acc[r] + lds_emb[mrow * DD + d0 + col];
        float s = val[r] * val[r];
        // reduce over the 16 lanes of this half (same row, 16 cols)
        s += __shfl_xor(s, 1, 32);
        s += __shfl_xor(s, 2, 32);
        s += __shfl_xor(s, 4, 32);
        s += __shfl_xor(s, 8, 32);
        if (col == 0) atomicAdd(&lds_ss[mrow], s);
    }
    __syncthreads();

    // ---- normalize and store ----
#pragma unroll
    for (int r = 0; r < 8; ++r) {
        const int mrow  = r + 8 * half;
        const float nrm = sqrtf(lds_ss[mrow]);
        const float scl = 1.0f / fmaxf(nrm, EPSF);
        out[((size_t)(b0 + mrow) * TT + t) * DD + d0 + col] = val[r] * scl;
    }
}

// ---------------------------------------------------------------------------
extern "C" void kernel_launch(void* const* d_in, const int* in_sizes, int n_in,
                              void* d_out, int out_size, void* d_ws, size_t ws_size,
                              hipStream_t stream)
{
    (void)in_sizes; (void)n_in; (void)out_size; (void)ws_size;

    const float* node_embeddings = (const float*)d_in[0];  // (NUM_NODES, D)
    const float* node_type_emb   = (const float*)d_in[1];  // (NUM_NODES, T, U)
    const float* trans_w         = (const float*)d_in[2];  // (T, U, D)
    const float* trans_w_s1      = (const float*)d_in[3];  // (T, U, A)
    const float* trans_w_s2      = (const float*)d_in[4];  // (T, A, 1)
    const int*   input_nodes     = (const int*)d_in[5];    // (N_SRC)
    const int*   output_nodes    = (const int*)d_in[6];    // (B)
    const int*   edge_src        = (const int*)d_in[7];    // (T, E)
    const int*   edge_dst        = (const int*)d_in[8];    // (T, E)
    float*       out             = (float*)d_out;          // (B, T, D)

    // workspace: agg (T*B*U f32 = 4MB) | combined (B*U f32 = 1MB)
    float* agg      = (float*)d_ws;
    float* combined = agg + (size_t)TT * BB * UU;

    hipMemsetAsync(agg, 0, (size_t)TT * BB * UU * sizeof(float), stream);

    dglgatne_scatter<<<dim3(EE / 8, TT), 256, 0, stream>>>(
        edge_src, edge_dst, input_nodes, node_type_emb, agg);

    dglgatne_attention<<<BB / 8, 256, 0, stream>>>(
        agg, trans_w_s1, trans_w_s2, combined);

    dglgatne_gemm_norm<<<dim3(BB / 16, TT), 256, 0, stream>>>(
        combined, trans_w, node_embeddings, output_nodes, out);
}